// CLR_Loss_84756884619446
// MI455X (gfx1250) — compile-verified
//
#include <hip/hip_runtime.h>

// ---------------------------------------------------------------------------
// CLRNet-style lane loss on gfx1250 (MI455X).
// Memory-bound: 92MB prediction stream @ 23.3TB/s ~= 4us floor; ~2.5 GFLOP is
// negligible. No GEMM structure (abs/min/max inside every K-contraction) =>
// WMMA inapplicable. CDNA5 paths used: async global->LDS DMA (b128, NT hint,
// ASYNCcnt), conflict-free LDS row reads, wave32 shuffle reductions.
// ---------------------------------------------------------------------------

#define IMG_W     800.0f
#define IMG_H     320.0f
#define NPTS      72
#define N_STRIPSF 71.0f
#define MAX_LANES 4
#define NUM_PRIORS 192
#define STAGES    3
#define BATCH     512
#define LIOU_LEN  15.0f
#define DD        78          // 6 + NPTS
#define CLS_W     2.0f
#define XYT_W     0.2f
#define IOU_W     2.0f
#define SEG_W     1.0f

__device__ __forceinline__ float smooth_l1f(float x) {
    float ax = fabsf(x);
    return ax < 1.0f ? 0.5f * x * x : ax - 0.5f;
}

// 192-thread (6 wave32) block reductions: wave shuffle + tiny LDS combine.
__device__ __forceinline__ float block_sum6(float v, volatile float* scr) {
#pragma unroll
    for (int off = 16; off > 0; off >>= 1) v += __shfl_xor(v, off, 32);
    __syncthreads();
    if ((threadIdx.x & 31) == 0) scr[threadIdx.x >> 5] = v;
    __syncthreads();
    float r = scr[0];
#pragma unroll
    for (int w = 1; w < 6; ++w) r += scr[w];
    return r;
}

__device__ __forceinline__ float block_max6(float v, volatile float* scr) {
#pragma unroll
    for (int off = 16; off > 0; off >>= 1) v = fmaxf(v, __shfl_xor(v, off, 32));
    __syncthreads();
    if ((threadIdx.x & 31) == 0) scr[threadIdx.x >> 5] = v;
    __syncthreads();
    float r = scr[0];
#pragma unroll
    for (int w = 1; w < 6; ++w) r = fmaxf(r, scr[w]);
    return r;
}

// CDNA5 async copy: 16B global -> LDS, ASYNCcnt-tracked, no VGPR data path.
// Low 32 bits of a flat pointer into LDS are the LDS byte address (ISA 10.2).
__device__ __forceinline__ void async_b128_nt(unsigned lds, unsigned long long ga) {
    asm volatile("global_load_async_to_lds_b128 %0, %1, off th:TH_LOAD_NT"
                 :: "v"(lds), "v"(ga) : "memory");
}
__device__ __forceinline__ void async_b128(unsigned lds, unsigned long long ga) {
    asm volatile("global_load_async_to_lds_b128 %0, %1, off"
                 :: "v"(lds), "v"(ga) : "memory");
}

extern "C" __global__ void __launch_bounds__(192)
clr_loss_main(const float* __restrict__ preds,   // (S,B,192,78)
              const float* __restrict__ tgts,    // (B,4,78)
              float* __restrict__ blk_out)       // (S*B)
{
    __shared__ __align__(16) float s_pred[NUM_PRIORS * DD];   // 59904 B
    __shared__ __align__(16) float s_tgt[MAX_LANES * DD];     //  1248 B
    __shared__ __align__(16) float s_buf[NUM_PRIORS * MAX_LANES]; // 3072 B (iou_sg, then cost)
    __shared__ float s_scr[8];
    __shared__ int   s_dynk[MAX_LANES];

    const int tid    = threadIdx.x;          // prior index p
    const int sample = blockIdx.x;           // s*B + b
    const int b      = sample % BATCH;

    const float* gp = preds + (size_t)sample * (NUM_PRIORS * DD);
    const float* gt = tgts  + (size_t)b      * (MAX_LANES * DD);

    // ---- stage sample tiles into LDS via async DMA (coalesced b128) ----
    {
        // targets first (small, reused by all 3 stages -> default RT policy)
        if (tid < (MAX_LANES * DD) / 4) {    // 78 chunks
            async_b128((unsigned)(uintptr_t)&s_tgt[tid * 4],
                       (unsigned long long)(uintptr_t)(gt + tid * 4));
        }
        // prediction tile: 3744 chunks = 192*19 + 96; uniform loop + one tail
        // (no per-iteration EXEC manipulation), streamed non-temporal.
        unsigned           lds = (unsigned)(uintptr_t)&s_pred[tid * 4];
        unsigned long long ga  = (unsigned long long)(uintptr_t)(gp + tid * 4);
#pragma unroll
        for (int j = 0; j < 19; ++j) {
            async_b128_nt(lds, ga);
            lds += 192 * 16;
            ga  += 192 * 16;
        }
        if (tid < 96) async_b128_nt(lds, ga);
        asm volatile("s_wait_asynccnt 0" ::: "memory");
    }
    __syncthreads();

    // ---- per-lane validity ----
    bool valid[MAX_LANES];
    int  num_t = 0;
#pragma unroll
    for (int l = 0; l < MAX_LANES; ++l) {
        valid[l] = (s_tgt[l * DD + 1] == 1.0f);
        num_t += valid[l] ? 1 : 0;
    }
    const bool has_t = num_t > 0;

    const float* P = &s_pred[tid * DD];   // stride-78 rows: bank-conflict-free

    // ---- pairwise line IoU + masked L1 distance over NPTS ----
    float ovr[MAX_LANES]  = {0, 0, 0, 0};
    float uni[MAX_LANES]  = {0, 0, 0, 0};
    float dsum[MAX_LANES] = {0, 0, 0, 0};
    float tlen[MAX_LANES] = {0, 0, 0, 0};
    for (int k = 0; k < NPTS; ++k) {
        float px = P[6 + k] * (IMG_W - 1.0f);
#pragma unroll
        for (int l = 0; l < MAX_LANES; ++l) {
            float t = s_tgt[l * DD + 6 + k];        // LDS broadcast read
            bool inv = (t < 0.0f) || (t >= IMG_W);
            if (!inv) {
                ovr[l]  += fminf(px + LIOU_LEN, t + LIOU_LEN) - fmaxf(px - LIOU_LEN, t - LIOU_LEN);
                uni[l]  += fmaxf(px + LIOU_LEN, t + LIOU_LEN) - fminf(px - LIOU_LEN, t - LIOU_LEN);
                dsum[l] += fabsf(t - px);
                tlen[l] += 1.0f;
            }
        }
    }
    float iou[MAX_LANES], dist[MAX_LANES];
#pragma unroll
    for (int l = 0; l < MAX_LANES; ++l) {
        iou[l]  = ovr[l]  / (uni[l]  + 1e-9f);
        dist[l] = dsum[l] / (tlen[l] + 1e-9f);
    }

    // ---- start-point / angle distances ----
    float sd[MAX_LANES], th[MAX_LANES];
    {
        float py  = P[2] * (IMG_H - 1.0f);
        float pxx = P[3] * (IMG_W - 1.0f);
        float pth = P[4];
#pragma unroll
        for (int l = 0; l < MAX_LANES; ++l) {
            float dy = py  - s_tgt[l * DD + 2] * (IMG_H - 1.0f);
            float dx = pxx - s_tgt[l * DD + 3];
            sd[l] = sqrtf(dy * dy + dx * dx);
            th[l] = fabsf(pth - s_tgt[l * DD + 4]) * 180.0f;
        }
    }

    // ---- global masked maxima for norm_score ----
    float mD = -__builtin_inff(), mS = -__builtin_inff(), mT = -__builtin_inff();
#pragma unroll
    for (int l = 0; l < MAX_LANES; ++l)
        if (valid[l]) {
            mD = fmaxf(mD, dist[l]);
            mS = fmaxf(mS, sd[l]);
            mT = fmaxf(mT, th[l]);
        }
    mD = block_max6(mD, s_scr);
    mS = block_max6(mS, s_scr);
    mT = block_max6(mT, s_scr);
    if (!has_t) { mD = 1.0f; mS = 1.0f; mT = 1.0f; }

    // ---- classification assignment cost (column 1 only) ----
    const float l0 = P[0], l1 = P[1];
    float ccost;
    {
        float sp1  = 1.0f / (1.0f + expf(-l1));
        float neg1 = -logf(1.0f - sp1 + 1e-12f) * 0.75f * sp1 * sp1;
        float pos1 = -logf(sp1 + 1e-12f) * 0.25f * (1.0f - sp1) * (1.0f - sp1);
        ccost = pos1 - neg1;
    }

    // ---- total matching cost ----
    float cost[MAX_LANES];
#pragma unroll
    for (int l = 0; l < MAX_LANES; ++l) {
        float dsc = 1.0f - dist[l] / mD + 0.01f;
        float ssc = 1.0f - sd[l]   / mS + 0.01f;
        float tsc = 1.0f - th[l]   / mT + 0.01f;
        float pr  = dsc * ssc * tsc;
        float c   = -(pr * pr) * 3.0f + ccost;
        cost[l] = valid[l] ? c : __builtin_inff();
    }

    // ---- phase A: dyn_k from top-4 of clamped IoU per lane ----
#pragma unroll
    for (int l = 0; l < MAX_LANES; ++l)
        s_buf[l * NUM_PRIORS + tid] = valid[l] ? fmaxf(iou[l], 0.0f) : 0.0f;
    __syncthreads();
    if (tid < MAX_LANES) {
        float a = -1.0f, bb = -1.0f, c = -1.0f, d = -1.0f;
        const float4* col = (const float4*)&s_buf[tid * NUM_PRIORS];
        for (int qq = 0; qq < NUM_PRIORS / 4; ++qq) {
            float4 v4 = col[qq];
            float vv[4] = {v4.x, v4.y, v4.z, v4.w};
#pragma unroll
            for (int c4 = 0; c4 < 4; ++c4) {
                float v = vv[c4];
                if (v > a)       { d = c; c = bb; bb = a; a = v; }
                else if (v > bb) { d = c; c = bb; bb = v; }
                else if (v > c)  { d = c; c = v; }
                else if (v > d)  { d = v; }
            }
        }
        int k = (int)(a + bb + c + d);
        s_dynk[tid] = k < 1 ? 1 : k;
    }
    __syncthreads();

    // ---- phase B: reuse buffer for cost columns, compute stable ranks ----
#pragma unroll
    for (int l = 0; l < MAX_LANES; ++l)
        s_buf[l * NUM_PRIORS + tid] = cost[l];
    __syncthreads();

    float Mf[MAX_LANES];
    int   msum = 0;
#pragma unroll
    for (int l = 0; l < MAX_LANES; ++l) {
        float c = cost[l];
        int r = 0;
        const float4* col = (const float4*)&s_buf[l * NUM_PRIORS];
        for (int qq = 0; qq < NUM_PRIORS / 4; ++qq) {
            float4 v4 = col[qq];                       // ds_load_b128 broadcast
            int qb = qq * 4;
            r += ((v4.x < c) || (v4.x == c && (qb + 0) < tid)) ? 1 : 0;
            r += ((v4.y < c) || (v4.y == c && (qb + 1) < tid)) ? 1 : 0;
            r += ((v4.z < c) || (v4.z == c && (qb + 2) < tid)) ? 1 : 0;
            r += ((v4.w < c) || (v4.w == c && (qb + 3) < tid)) ? 1 : 0;
        }
        bool M = (r < s_dynk[l]) && valid[l];
        Mf[l] = M ? 1.0f : 0.0f;
        msum += M ? 1 : 0;
    }
    if (msum > 1) {   // multi-match: clear lane 0, set argmin-cost lane (jnp.argmin = first min)
        int amin = 0;
        float cmin = cost[0];
#pragma unroll
        for (int l = 1; l < MAX_LANES; ++l)
            if (cost[l] < cmin) { cmin = cost[l]; amin = l; }
        Mf[0] = 0.0f;
#pragma unroll
        for (int l = 0; l < MAX_LANES; ++l)
            if (l == amin) Mf[l] = 1.0f;
    }

    // ---- per-prior loss partials ----
    float nm = Mf[0] + Mf[1] + Mf[2] + Mf[3];

    float cls_part;
    {
        float mx = fmaxf(l0, l1);
        float e0 = expf(l0 - mx), e1 = expf(l1 - mx);
        float invs = 1.0f / (e0 + e1);
        float p0 = e0 * invs + 1e-8f, p1 = e1 * invs + 1e-8f;
        float f0 = -0.25f * (1.0f - p0) * (1.0f - p0) * logf(p0);
        float f1 = -0.25f * (1.0f - p1) * (1.0f - p1) * logf(p1);
        cls_part = (nm > 0.0f) ? f1 : f0;
    }

    float y0 = P[2] * N_STRIPSF;
    float y1 = P[3] * (IMG_W - 1.0f);
    float y2 = P[4] * 180.0f;
    float y3 = P[5] * N_STRIPSF;
    float pstart = fminf(fmaxf(rintf(P[2] * N_STRIPSF), 0.0f), N_STRIPSF);

    float reg_part = 0.0f, iou_part = 0.0f;
#pragma unroll
    for (int l = 0; l < MAX_LANES; ++l) {
        float t2 = s_tgt[l * DD + 2];
        float tstart = rintf(t2 * N_STRIPSF);
        float tl = s_tgt[l * DD + 5] - (pstart - tstart);
        float sl = smooth_l1f(y0 - t2 * N_STRIPSF)
                 + smooth_l1f(y1 - s_tgt[l * DD + 3])
                 + smooth_l1f(y2 - s_tgt[l * DD + 4] * 180.0f)
                 + smooth_l1f(y3 - tl);
        reg_part += Mf[l] * sl;
        iou_part += Mf[l] * (1.0f - iou[l]);
    }

    // ---- block reductions + per-sample contribution ----
    float nm_tot  = block_sum6(nm,       s_scr);
    float cls_tot = block_sum6(cls_part, s_scr);
    float reg_tot = block_sum6(reg_part, s_scr);
    float iou_tot = block_sum6(iou_part, s_scr);

    if (tid == 0) {
        float cls_term = cls_tot / (has_t ? (float)num_t : 1.0f);
        float reg_term = has_t ? reg_tot / fmaxf(nm_tot * 4.0f, 1.0f) : 0.0f;
        float iou_term = has_t ? iou_tot / fmaxf(nm_tot, 1.0f) : 0.0f;
        blk_out[blockIdx.x] = (cls_term * CLS_W + reg_term * XYT_W + iou_term * IOU_W)
                              * (1.0f / (float)(BATCH * STAGES));
    }
}

// Deterministic final reduction (fixed tree) + seg_loss term.
extern "C" __global__ void __launch_bounds__(256)
clr_loss_final(const float* __restrict__ blk, const float* __restrict__ seg,
               float* __restrict__ out, int n)
{
    __shared__ float s[256];
    float v = 0.0f;
    for (int i = threadIdx.x; i < n; i += 256) v += blk[i];
    s[threadIdx.x] = v;
    __syncthreads();
#pragma unroll
    for (int off = 128; off > 0; off >>= 1) {
        if (threadIdx.x < off) s[threadIdx.x] += s[threadIdx.x + off];
        __syncthreads();
    }
    if (threadIdx.x == 0) out[0] = s[0] + seg[0] * SEG_W;
}

extern "C" void kernel_launch(void* const* d_in, const int* in_sizes, int n_in,
                              void* d_out, int out_size, void* d_ws, size_t ws_size,
                              hipStream_t stream) {
    (void)in_sizes; (void)n_in; (void)out_size; (void)ws_size;
    const float* preds = (const float*)d_in[0];   // (3,512,192,78) f32
    const float* tgts  = (const float*)d_in[1];   // (512,4,78)     f32
    const float* seg   = (const float*)d_in[2];   // scalar         f32
    float* out = (float*)d_out;
    float* blk = (float*)d_ws;                    // 1536 f32 partials

    const int nblocks = STAGES * BATCH;           // 1536 samples
    clr_loss_main<<<nblocks, 192, 0, stream>>>(preds, tgts, blk);
    clr_loss_final<<<1, 256, 0, stream>>>(blk, seg, out, nblocks);
}